// PerceptronSP_12678743458400
// MI455X (gfx1250) — compile-verified
//
#include <hip/hip_runtime.h>

#define DIM_IN 25
#define HN     128
#define NPIX   8000
#define BSZ    32
#define NEGSL  0.01f

typedef __attribute__((ext_vector_type(2))) float v2f;
typedef __attribute__((ext_vector_type(8))) float v8f;

// LDS strides (floats) chosen for bank-conflict-free fragment access (64 banks)
#define XS_S 28    // x tile: 32 x 28 (zero padded cols 25..27)
#define H1_S 132   // h1 tile: 32 rows, 4*row%64 spreads banks; +2 col offset hits odd banks
#define WB_S 144   // weight tile: 16*row%64 -> half-wave K-pairs land 32 banks apart

__device__ __forceinline__ float leaky(float v) { return v >= 0.f ? v : NEGSL * v; }

__global__ __launch_bounds__(256)
void pixel_mlp_wmma(const float* __restrict__ x,
                    const float* __restrict__ W1,
                    const float* __restrict__ b1,
                    const float* __restrict__ W2,
                    const float* __restrict__ b2,
                    const float* __restrict__ W3,
                    const float* __restrict__ b3,
                    float* __restrict__ y)
{
    __shared__ float xs[BSZ * XS_S];     // staged x, zero-padded to K=28
    __shared__ float h1s[BSZ * H1_S];    // layer-1 activations 32x128
    __shared__ float wbuf[32 * WB_S];    // staged W1 (25 rows) / W2 chunk (32 rows)
    __shared__ float yacc[BSZ];

    const int p    = blockIdx.x;
    const int tid  = threadIdx.x;
    const int lane = tid & 31;           // wave32
    const int wave = tid >> 5;           // 8 waves: wave = N-tile (16 output cols)
    const int hi   = (lane >= 16) ? 1 : 0;
    const int l16  = lane & 15;
    const int ncol = wave * 16;

    const float* w1p = W1 + (size_t)p * DIM_IN * HN;
    const float* w2p = W2 + (size_t)p * HN * HN;
    const float* b1p = b1 + (size_t)p * HN;
    const float* b2p = b2 + (size_t)p * HN;
    const float* w3p = W3 + (size_t)p * HN;

    // ---- stage x (zero-padded), W1 (coalesced float4), init yacc ----
    for (int i = tid; i < BSZ * XS_S; i += 256) {
        int r = i / XS_S, c = i - r * XS_S;
        xs[i] = (c < DIM_IN) ? x[r * DIM_IN + c] : 0.f;
    }
    {
        const float4* w1p4 = (const float4*)w1p;
        for (int j = tid; j < (DIM_IN * HN) / 4; j += 256) {
            int r = j >> 5, c4 = (j & 31) << 2;
            *(float4*)&wbuf[r * WB_S + c4] = w1p4[j];
        }
        for (int i = tid; i < 3 * HN; i += 256) {           // zero K=25..27
            int r = DIM_IN + i / HN, c = i % HN;
            wbuf[r * WB_S + c] = 0.f;
        }
    }
    if (tid < BSZ) yacc[tid] = 0.f;
    // prefetch this pixel's W2 (64 KB) while layer 1 runs
    for (int i = tid; i < (HN * HN) / 32; i += 256)
        __builtin_prefetch(w2p + i * 32, 0, 1);
    __syncthreads();

    // ---- layer 1: [32x25(28)] x [25(28)x128] via f32 WMMA, K-steps of 4 ----
    v8f c0 = {0.f,0.f,0.f,0.f,0.f,0.f,0.f,0.f};
    v8f c1 = c0;
    #pragma unroll
    for (int kk = 0; kk < 7; ++kk) {
        const int kb = 4 * kk + 2 * hi;                 // this half-wave's K pair
        v2f b;                                          // B: 4x16, VGPR0=K0/K2, VGPR1=K1/K3
        b.x = wbuf[(kb + 0) * WB_S + ncol + l16];
        b.y = wbuf[(kb + 1) * WB_S + ncol + l16];
        v2f a0, a1;                                     // A: 16x4, M=lane%16
        a0.x = xs[l16 * XS_S + kb];        a0.y = xs[l16 * XS_S + kb + 1];
        a1.x = xs[(16 + l16) * XS_S + kb]; a1.y = xs[(16 + l16) * XS_S + kb + 1];
        c0 = __builtin_amdgcn_wmma_f32_16x16x4_f32(false, a0, false, b, (short)0, c0, false, false);
        c1 = __builtin_amdgcn_wmma_f32_16x16x4_f32(false, a1, false, b, (short)0, c1, false, false);
    }
    {   // bias + leaky, scatter D-frags to LDS (D: VGPR v -> row v + 8*hi, col = lane%16)
        const float bb = b1p[ncol + l16];
        #pragma unroll
        for (int v = 0; v < 8; ++v) {
            const int r0 = v + 8 * hi;
            h1s[r0 * H1_S + ncol + l16]        = leaky(c0[v] + bb);
            h1s[(16 + r0) * H1_S + ncol + l16] = leaky(c1[v] + bb);
        }
    }
    __syncthreads();   // h1 complete; wbuf (W1) reads done

    // ---- layer 2: [32x128] x [128x128], W2 streamed in 4 chunks of 32 K-rows ----
    v8f d0 = {0.f,0.f,0.f,0.f,0.f,0.f,0.f,0.f};
    v8f d1 = d0;
    const float4* w2p4 = (const float4*)w2p;
    for (int chunk = 0; chunk < 4; ++chunk) {
        if (chunk) __syncthreads();                     // prior chunk reads done
        for (int j = tid; j < 1024; j += 256) {         // 32 rows x 128 f, coalesced b128
            int r = j >> 5, c4 = (j & 31) << 2;
            *(float4*)&wbuf[r * WB_S + c4] = w2p4[chunk * 1024 + j];
        }
        __syncthreads();
        const int kbase = chunk * 32;
        #pragma unroll
        for (int kk = 0; kk < 8; ++kk) {
            const int kb = 4 * kk + 2 * hi;
            v2f b;
            b.x = wbuf[(kb + 0) * WB_S + ncol + l16];
            b.y = wbuf[(kb + 1) * WB_S + ncol + l16];
            v2f a0, a1;
            const int hc = kbase + kb;
            a0.x = h1s[l16 * H1_S + hc];        a0.y = h1s[l16 * H1_S + hc + 1];
            a1.x = h1s[(16 + l16) * H1_S + hc]; a1.y = h1s[(16 + l16) * H1_S + hc + 1];
            d0 = __builtin_amdgcn_wmma_f32_16x16x4_f32(false, a0, false, b, (short)0, d0, false, false);
            d1 = __builtin_amdgcn_wmma_f32_16x16x4_f32(false, a1, false, b, (short)0, d1, false, false);
        }
    }

    // ---- layer 3: leaky(h2 + b2) . W3 reduced across columns via LDS float atomics ----
    {
        const float bb = b2p[ncol + l16];
        const float wc = w3p[ncol + l16];
        #pragma unroll
        for (int v = 0; v < 8; ++v) {
            const int r0 = v + 8 * hi;
            atomicAdd(&yacc[r0],      leaky(d0[v] + bb) * wc);
            atomicAdd(&yacc[16 + r0], leaky(d1[v] + bb) * wc);
        }
    }
    __syncthreads();
    if (tid < BSZ) y[tid * NPIX + p] = yacc[tid] + b3[p];
}

extern "C" void kernel_launch(void* const* d_in, const int* in_sizes, int n_in,
                              void* d_out, int out_size, void* d_ws, size_t ws_size,
                              hipStream_t stream) {
    const float* x  = (const float*)d_in[0];
    const float* W1 = (const float*)d_in[1];
    const float* b1 = (const float*)d_in[2];
    const float* W2 = (const float*)d_in[3];
    const float* b2 = (const float*)d_in[4];
    const float* W3 = (const float*)d_in[5];
    const float* b3 = (const float*)d_in[6];
    float* y = (float*)d_out;
    pixel_mlp_wmma<<<dim3(NPIX), dim3(256), 0, stream>>>(x, W1, b1, W2, b2, W3, b3, y);
}